// TabNetEncoder_49297634624105
// MI455X (gfx1250) — compile-verified
//
#include <hip/hip_runtime.h>
#include <hip/hip_bf16.h>
#include <math.h>

#define BATCH 16384
#define DIMD  256
#define NTF   256
#define NF    128
#define VB    512
#define KC    16
#define BCOL  128          // columns per block (half of NTF)
#define ASTRIDE 20         // dwords per LDS A row (16 data + 4 pad -> conflict-free)
#define BSTRIDE 136        // dwords per LDS B row (128 data + 8 pad)

#define MODE_GLU  0
#define MODE_GBN  1
#define MODE_MASK 2

typedef float v2f __attribute__((ext_vector_type(2)));
typedef float v8f __attribute__((ext_vector_type(8)));

__device__ __forceinline__ float wave_sum(float v) {
#pragma unroll
  for (int o = 16; o > 0; o >>= 1) v += __shfl_xor(v, o, 32);
  return v;
}
__device__ __forceinline__ float wave_max(float v) {
#pragma unroll
  for (int o = 16; o > 0; o >>= 1) v = fmaxf(v, __shfl_xor(v, o, 32));
  return v;
}

// CDNA5 async global->LDS copy (ASYNCcnt-tracked, no VGPR round-trip).
__device__ __forceinline__ void async_b128(unsigned lds_off, const void* gaddr) {
  asm volatile("global_load_async_to_lds_b128 %0, %1, off"
               :: "v"(lds_off), "v"((unsigned long long)(uintptr_t)gaddr)
               : "memory");
}
__device__ __forceinline__ void wait_async0() {
  asm volatile("s_wait_asynccnt 0x0" ::: "memory");
}

// ---------------------------------------------------------------- init
__global__ void init_kernel(float* __restrict__ stats, float* __restrict__ prior) {
  long long i = (long long)blockIdx.x * blockDim.x + threadIdx.x;
  const long long n = (long long)BATCH * DIMD;
  const long long stride = (long long)gridDim.x * blockDim.x;
  for (; i < n; i += stride) {
    if (i < 1024) stats[i] = 0.f;
    prior[i] = 1.f;
  }
}

// ------------------------------------------------------- full-batch BN
__global__ void bn_stats_kernel(const float* __restrict__ x, float* __restrict__ stats) {
  const int c = threadIdx.x;            // column 0..255
  const int r0 = blockIdx.x * 128;
  float s = 0.f, sq = 0.f;
  for (int r = r0; r < r0 + 128; ++r) {
    float v = x[(long long)r * DIMD + c];
    s += v; sq += v * v;
  }
  atomicAdd(&stats[c], s);
  atomicAdd(&stats[DIMD + c], sq);
}

__global__ void bn_apply_kernel(const float* __restrict__ x, const float* __restrict__ stats,
                                const float* __restrict__ w, const float* __restrict__ bb,
                                float* __restrict__ xn) {
  long long i = (long long)blockIdx.x * blockDim.x + threadIdx.x;
  const long long n = (long long)BATCH * DIMD;
  const long long stride = (long long)gridDim.x * blockDim.x;
  for (; i < n; i += stride) {
    int c = (int)(i & (DIMD - 1));
    float mean = stats[c] * (1.f / BATCH);
    float var  = stats[DIMD + c] * (1.f / BATCH) - mean * mean;
    xn[i] = (x[i] - mean) * rsqrtf(var + 1e-5f) * w[c] + bb[c];
  }
}

// ------------------------------------------------ fused GEMM + GBN + epilogue
// Block = 512 rows x 128 cols of C = A(512xK) @ W(K x 256) (column half cb).
// 512 threads = 16 waves (8 row-pos x 2 col-pos); wave owns 64x64 = 16 WMMA
// f32 16x16x4 tiles = 128 acc VGPRs (4 waves/SIMD, 256-VGPR budget, no spill).
// K is streamed in KC=16 chunks, double-buffered in LDS and filled with
// GLOBAL_LOAD_ASYNC_TO_LDS_B128: chunk i+1 streams in while WMMAs consume
// chunk i; s_wait_asynccnt 0 + barrier closes each iteration.
// Ghost-BN (512-row virtual batches) stays block-local (column split).
// Block cb owns global cols {cb*64..+63} u {128+cb*64..+63}; wave wc owns
// local tiles {2wc,2wc+1,2wc+4,2wc+5} so the GLU gate partner (c, c+128)
// lives in the same lane & VGPR slot.
template <int MODE>
__global__ __launch_bounds__(512, 1)
void gemm512_kernel(const float* __restrict__ A, int lda, int K,
                    const float* __restrict__ W,
                    const float* __restrict__ g, const float* __restrict__ b,
                    const float* __restrict__ resid,  // GLU residual (B x 128), may be null
                    const float* __restrict__ xmul,   // MASK elementwise (B x 256)
                    float* __restrict__ out) {
  __shared__ float As2[2][VB * ASTRIDE];    // row-major A, 2 x 40 KB
  __shared__ float Bs2[2][KC * BSTRIDE];    // 2 x 8.5 KB
  __shared__ float red[2][BCOL];            // colsum / colsumsq for ghost-BN
  __shared__ float ssc[BCOL];
  __shared__ float ssh[BCOL];

  const int tid  = threadIdx.x;
  const int lane = tid & 31;
  const int w    = tid >> 5;          // 0..15
  const int wr   = w & 7;             // row position (64-row stripes)
  const int wc   = w >> 3;            // 0..1 col position
  const int half = lane >> 4;         // 0: K=0,1 / M=0..7 ; 1: K=2,3 / M=8..15
  const int ln   = lane & 15;
  const int rowBase = wr * 64;
  const int rb = blockIdx.x >> 1;
  const int cb = blockIdx.x & 1;
  const long long gRow0 = (long long)rb * VB;

  const unsigned ldsA = (unsigned)(uintptr_t)&As2[0][0];  // LDS offset = addr[31:0]
  const unsigned ldsB = (unsigned)(uintptr_t)&Bs2[0][0];

  int ltile[4];   // local col-tile indices (each *16 -> local col)
  ltile[0] = 2 * wc;     ltile[1] = 2 * wc + 1;
  ltile[2] = 2 * wc + 4; ltile[3] = 2 * wc + 5;
  int gcol0[4];   // global col base per tile
#pragma unroll
  for (int t = 0; t < 4; ++t)
    gcol0[t] = (ltile[t] < 4) ? (cb * 64 + ltile[t] * 16)
                              : (NF + cb * 64 + (ltile[t] - 4) * 16);

  // Per-thread staging coordinates (fixed across chunks).
  const int aRow = tid >> 2;           // with 4 passes: rows tid>>2 + s*128
  const int aKq  = tid & 3;
  const int bK   = tid >> 5;
  const int bN4  = tid & 31;
  const int bGc  = (bN4 < 16) ? (cb * 64 + bN4 * 4) : (NF + cb * 64 + (bN4 - 16) * 4);
  const int bLc  = bN4 * 4;

  // Issue async fills of LDS buffer `sel` with K-chunk starting at k0.
  auto stage = [&](int k0, int sel) {
    unsigned aBase = ldsA + (unsigned)sel * (VB * ASTRIDE * 4);
    unsigned bBase = ldsB + (unsigned)sel * (KC * BSTRIDE * 4);
#pragma unroll
    for (int s = 0; s < 4; ++s) {
      int row = aRow + s * 128;
      async_b128(aBase + (unsigned)(row * ASTRIDE + aKq * 4) * 4,
                 &A[(gRow0 + row) * lda + k0 + aKq * 4]);
    }
    async_b128(bBase + (unsigned)(bK * BSTRIDE + bLc) * 4,
               &W[(long long)(k0 + bK) * NTF + bGc]);
  };

  v8f acc[4][4];
#pragma unroll
  for (int i = 0; i < 4; ++i)
#pragma unroll
    for (int j = 0; j < 4; ++j)
#pragma unroll
      for (int v = 0; v < 8; ++v) acc[i][j][v] = 0.f;

  const int nchunks = K / KC;
  stage(0, 0);
  wait_async0();
  __syncthreads();

  for (int i = 0; i < nchunks; ++i) {
    const int cur = i & 1;
    if (i + 1 < nchunks) stage((i + 1) * KC, (i + 1) & 1);  // overlap with compute

    const float* Ab = &As2[cur][0];
    const float* Bb = &Bs2[cur][0];
#pragma unroll
    for (int kk = 0; kk < 4; ++kk) {
      const int kb = kk * 4 + half * 2;
      v2f af[4], bf[4];
#pragma unroll
      for (int rt = 0; rt < 4; ++rt) {
        int r = rowBase + rt * 16 + ln;
        af[rt][0] = Ab[r * ASTRIDE + kb];      // contiguous pair -> ds_load_b64
        af[rt][1] = Ab[r * ASTRIDE + kb + 1];
      }
#pragma unroll
      for (int ct = 0; ct < 4; ++ct) {
        int c = ltile[ct] * 16 + ln;
        bf[ct][0] = Bb[kb * BSTRIDE + c];
        bf[ct][1] = Bb[(kb + 1) * BSTRIDE + c];
      }
#pragma unroll
      for (int rt = 0; rt < 4; ++rt)
#pragma unroll
        for (int ct = 0; ct < 4; ++ct)
          acc[rt][ct] = __builtin_amdgcn_wmma_f32_16x16x4_f32(
              false, af[rt], false, bf[ct], (short)0, acc[rt][ct], false, false);
    }
    wait_async0();     // next chunk resident (ran during compute)
    __syncthreads();
  }

  if (MODE == MODE_MASK) {
    // masked_x = (M @ group_mat) * x, no normalization
#pragma unroll
    for (int rt = 0; rt < 4; ++rt)
#pragma unroll
      for (int ct = 0; ct < 4; ++ct) {
        int gc = gcol0[ct] + ln;
#pragma unroll
        for (int v = 0; v < 8; ++v) {
          long long row = gRow0 + rowBase + rt * 16 + half * 8 + v;
          out[row * NTF + gc] = acc[rt][ct][v] * xmul[row * NTF + gc];
        }
      }
    return;
  }

  // Ghost-BN: block covers a full 512-row virtual batch for its columns.
  for (int i = tid; i < 2 * BCOL; i += 512) (&red[0][0])[i] = 0.f;
  __syncthreads();
#pragma unroll
  for (int ct = 0; ct < 4; ++ct) {
    float s = 0.f, sq = 0.f;
#pragma unroll
    for (int rt = 0; rt < 4; ++rt)
#pragma unroll
      for (int v = 0; v < 8; ++v) { float t = acc[rt][ct][v]; s += t; sq += t * t; }
    int lc = ltile[ct] * 16 + ln;
    atomicAdd(&red[0][lc], s);
    atomicAdd(&red[1][lc], sq);
  }
  __syncthreads();
  if (tid < BCOL) {
    int lc = tid;
    int gc = (lc < 64) ? (cb * 64 + lc) : (NF + cb * 64 + (lc - 64));
    float mean = red[0][lc] * (1.f / VB);
    float var  = red[1][lc] * (1.f / VB) - mean * mean;
    float rstd = rsqrtf(var + 1e-5f);
    float sc   = rstd * g[gc];
    ssc[lc] = sc;
    ssh[lc] = b[gc] - mean * sc;
  }
  __syncthreads();

  if (MODE == MODE_GBN) {
#pragma unroll
    for (int ct = 0; ct < 4; ++ct) {
      int lc = ltile[ct] * 16 + ln;
      int gc = gcol0[ct] + ln;
      float sc = ssc[lc], sh = ssh[lc];
#pragma unroll
      for (int rt = 0; rt < 4; ++rt)
#pragma unroll
        for (int v = 0; v < 8; ++v) {
          long long row = gRow0 + rowBase + rt * 16 + half * 8 + v;
          out[row * NTF + gc] = acc[rt][ct][v] * sc + sh;
        }
    }
  } else {  // MODE_GLU: h[:, :128] * sigmoid(h[:, 128:]), optional residual * sqrt(0.5)
#pragma unroll
    for (int j = 0; j < 2; ++j) {
      int lcA = ltile[j] * 16 + ln;
      int lcB = ltile[j + 2] * 16 + ln;
      int cA  = gcol0[j] + ln;        // 0..127 ; partner gcol0[j+2]+ln == cA+128
      float scA = ssc[lcA], shA = ssh[lcA], scB = ssc[lcB], shB = ssh[lcB];
#pragma unroll
      for (int rt = 0; rt < 4; ++rt)
#pragma unroll
        for (int v = 0; v < 8; ++v) {
          long long row = gRow0 + rowBase + rt * 16 + half * 8 + v;
          float ha = acc[rt][j][v]     * scA + shA;
          float hb = acc[rt][j + 2][v] * scB + shB;
          float gl = ha * (1.f / (1.f + __expf(-hb)));
          if (resid) gl = (resid[row * NF + cA] + gl) * 0.70710678118654752f;
          out[row * NF + cA] = gl;
        }
    }
  }
}

// -------------------------------------------------- sparsemax (bisection on tau)
// One wave32 per row of 256. tau solves sum(max(z-tau,0))=1, monotone -> bisect
// in [max(z)-1, max(z)]. Fuses loss accumulation and prior update.
__global__ void sparsemax_kernel(float* __restrict__ h /* in: logits, out: M */,
                                 float* __restrict__ prior, float* __restrict__ loss) {
  const int lane = threadIdx.x & 31;
  const long long row = (long long)blockIdx.x * 8 + (threadIdx.x >> 5);
  float z[8], pr[8];
#pragma unroll
  for (int i = 0; i < 8; ++i) {
    long long idx = row * 256 + lane + i * 32;
    pr[i] = prior[idx];
    z[i]  = h[idx] * pr[i];
  }
  float zmax = -3.4e38f;
#pragma unroll
  for (int i = 0; i < 8; ++i) zmax = fmaxf(zmax, z[i]);
  zmax = wave_max(zmax);
  float lo = zmax - 1.f, hi = zmax;
  for (int it = 0; it < 48; ++it) {
    float tau = 0.5f * (lo + hi);
    float s = 0.f;
#pragma unroll
    for (int i = 0; i < 8; ++i) s += fmaxf(z[i] - tau, 0.f);
    s = wave_sum(s);
    if (s > 1.f) lo = tau; else hi = tau;
  }
  float tau = 0.5f * (lo + hi);
  float ls = 0.f;
#pragma unroll
  for (int i = 0; i < 8; ++i) {
    long long idx = row * 256 + lane + i * 32;
    float m = fmaxf(z[i] - tau, 0.f);
    h[idx] = m;                        // M (in-place, row private to this wave)
    prior[idx] = (1.3f - m) * pr[i];   // GAMMA_RELAX = 1.3
    ls += m * __logf(m + 1e-15f);
  }
  ls = wave_sum(ls);
  if (lane == 0) atomicAdd(loss, ls);
}

// -------------------------------------------------------------- outputs
__global__ void relu_out_kernel(const float* __restrict__ h, float* __restrict__ out) {
  long long i = (long long)blockIdx.x * blockDim.x + threadIdx.x;
  const long long n = (long long)BATCH * 64;
  const long long stride = (long long)gridDim.x * blockDim.x;
  for (; i < n; i += stride)
    out[i] = fmaxf(h[(i >> 6) * NF + (i & 63)], 0.f);
}

__global__ void finalize_kernel(const float* __restrict__ loss, float* __restrict__ out) {
  out[0] = loss[0] * (1.f / (3.f * (float)BATCH));
}

// ---------------------------------------------------------------- host
extern "C" void kernel_launch(void* const* d_in, const int* in_sizes, int n_in,
                              void* d_out, int out_size, void* d_ws, size_t ws_size,
                              hipStream_t stream) {
  const float* x    = (const float*)d_in[0];
  const float* gmat = (const float*)d_in[1];
  const float* bn0w = (const float*)d_in[2];
  const float* bn0b = (const float*)d_in[3];
  const float* sW0  = (const float*)d_in[4];
  const float* sW1  = (const float*)d_in[5];
  const float* shg  = (const float*)d_in[6];
  const float* shb  = (const float*)d_in[7];
  const float* indW = (const float*)d_in[8];
  const float* indg = (const float*)d_in[9];
  const float* indb = (const float*)d_in[10];
  const float* attW = (const float*)d_in[11];
  const float* attg = (const float*)d_in[12];
  const float* attb = (const float*)d_in[13];
  float* out = (float*)d_out;

  float* ws    = (float*)d_ws;
  float* stats = ws;                                   // 512 sums + loss slot
  float* lossp = ws + 512;
  float* xn    = ws + 1024;                            // B x 256
  float* hA    = xn   + (long long)BATCH * 256;        // B x 128
  float* hB    = hA   + (long long)BATCH * 128;        // B x 128
  float* attn  = hB   + (long long)BATCH * 128;        // B x 256 (reused as M)
  float* prior = attn + (long long)BATCH * 256;        // B x 256
  float* maskd = prior+ (long long)BATCH * 256;        // B x 256

  init_kernel<<<512, 256, 0, stream>>>(stats, prior);
  bn_stats_kernel<<<BATCH / 128, 256, 0, stream>>>(x, stats);
  bn_apply_kernel<<<2048, 256, 0, stream>>>(x, stats, bn0w, bn0b, xn);

  const int GB = (BATCH / VB) * 2;   // 64 blocks: 32 row-groups x 2 col-halves

  auto ft = [&](const float* in, int lda, int f) {
    const float* g0 = shg + (f * 2 + 0) * 256, *b0 = shb + (f * 2 + 0) * 256;
    const float* g1 = shg + (f * 2 + 1) * 256, *b1 = shb + (f * 2 + 1) * 256;
    const float* w2 = indW + (long long)(f * 2 + 0) * 128 * 256;
    const float* w3 = indW + (long long)(f * 2 + 1) * 128 * 256;
    const float* g2 = indg + (f * 2 + 0) * 256, *bb2 = indb + (f * 2 + 0) * 256;
    const float* g3 = indg + (f * 2 + 1) * 256, *bb3 = indb + (f * 2 + 1) * 256;
    gemm512_kernel<MODE_GLU><<<GB, 512, 0, stream>>>(in, lda, 256, sW0, g0, b0, nullptr, nullptr, hA);
    gemm512_kernel<MODE_GLU><<<GB, 512, 0, stream>>>(hA, 128, 128, sW1, g1, b1, hA, nullptr, hB);
    gemm512_kernel<MODE_GLU><<<GB, 512, 0, stream>>>(hB, 128, 128, w2, g2, bb2, hB, nullptr, hA);
    gemm512_kernel<MODE_GLU><<<GB, 512, 0, stream>>>(hA, 128, 128, w3, g3, bb3, hA, nullptr, hB);
  };

  ft(xn, 256, 0);  // att lives in hB[:, 64:]
  for (int s = 0; s < 3; ++s) {
    gemm512_kernel<MODE_GBN><<<GB, 512, 0, stream>>>(hB + 64, 128, 64,
        attW + (long long)s * 64 * 256, attg + s * 256, attb + s * 256,
        nullptr, nullptr, attn);
    sparsemax_kernel<<<BATCH / 8, 256, 0, stream>>>(attn, prior, lossp);
    gemm512_kernel<MODE_MASK><<<GB, 512, 0, stream>>>(attn, 256, 256, gmat,
        nullptr, nullptr, nullptr, xn, maskd);
    ft(maskd, 256, s + 1);
    relu_out_kernel<<<1024, 256, 0, stream>>>(hB, out + (long long)s * BATCH * 64);
  }
  finalize_kernel<<<1, 1, 0, stream>>>(lossp, out + (long long)3 * BATCH * 64);
}